// Quantizer_23141283791676
// MI455X (gfx1250) — compile-verified
//
#include <hip/hip_runtime.h>

// MI455X / gfx1250 (CDNA5, wave32). Output = palette[argmax_k(logits+gumbel)].
// Logits via V_WMMA_F32_16X16X4_F32, C=64 reduction split into two independent
// 8-deep accumulator chains. Memory-bound: ~665 MB -> ~28.5us @ 23.3 TB/s.
// This revision uses unsigned 32-bit lane offsets so loads can lower to the
// saddr + voffset form (uniform SGPR base, one shared voffset VGPR), removing
// the residual v_add_co_u32/v_add_co_ci_u32 64-bit address pairs.

typedef float v2f __attribute__((ext_vector_type(2)));
typedef float v4f __attribute__((ext_vector_type(4)));
typedef float v8f __attribute__((ext_vector_type(8)));

#define HW    (512u * 512u)
#define WDIM  512
#define CDIM  64
#define KDIM  16
#define TPW   8              // 16-pixel WMMA tiles per wave (128 px/wave)
#define EPSF  1e-20f
#define LN2F  0.69314718055994530942f

__global__ __launch_bounds__(256)
void quantizer_wmma_kernel(const float* __restrict__ x,        // [8,64,512,512]
                           const float* __restrict__ conv_w,   // [16,64]
                           const float* __restrict__ conv_b,   // [16]
                           const float* __restrict__ palette,  // [16,3]
                           const float* __restrict__ unif,     // [8,512,512,16]
                           float* __restrict__ out)            // [8,3,512,512]
{
    const int lane = threadIdx.x & 31;

    // gwave is wave-uniform by construction; readfirstlane forces it (and all
    // derived bases) into SGPRs so loads lower to saddr + voffset form.
    const int gwave = __builtin_amdgcn_readfirstlane(
        (int)(blockIdx.x * 8u + (threadIdx.x >> 5)));

    const int p0  = gwave * (16 * TPW);   // 128 contiguous pixels, never crosses a row
    const int b   = p0 >> 18;             // / (512*512)
    const int rem = p0 & (int)(HW - 1);
    const int h   = rem >> 9;
    const int w0  = rem & (WDIM - 1);

    const uint32_t pl = (uint32_t)(lane & 15);  // pixel-in-tile (N) == k row (M)
    const uint32_t hi = (uint32_t)(lane >> 4);  // half-wave select
    const uint32_t dl = hi * 2u;                // A/B fragment channel delta
    const uint32_t ko = hi * 8u;                // C/D rows M = ko..ko+7

    // Wave-uniform (SGPR) row bases.
    const float* xrow = x    + (size_t)b * CDIM * HW + (size_t)h * WDIM + w0;
    const float* urow = unif + ((size_t)((b * 512 + h) * 512 + w0)) * KDIM;
    float*       orow = out  + (size_t)b * 3 * HW + (size_t)h * WDIM + w0;

    // Per-lane UNSIGNED element offsets, invariant over t and j: zext(i32)
    // pattern lets the backend fold them as the voffset of saddr-form loads.
    const uint32_t xoff = dl * HW + pl;         // x: channel-pair select + pixel
    const uint32_t uoff = pl * KDIM + ko;       // unif: K innermost, 16B aligned

    // A fragments: lane holds conv_w[k=pl][4j+dl], [4j+dl+1]; loaded once.
    v2f wa[16];
#pragma unroll
    for (int j = 0; j < 16; ++j) {
        const float* wp = conv_w + pl * CDIM + 4 * j + dl;
        wa[j].x = wp[0];
        wa[j].y = wp[1];
    }

    // Bias folded into chain-0 accumulator; chain 1 starts at zero.
    v8f cinit;
#pragma unroll
    for (int r = 0; r < 8; ++r) cinit[r] = conv_b[ko + r];

    for (int t = 0; t < TPW; ++t) {
        const float* xt = xrow + t * 16;          // scalar advance (SALU)

        // logits tile: D[16k x 16px] = conv_w(16x64) * x(64x16) + bias.
        // Two independent accumulator chains (depth 8 each) to relax the
        // WMMA D->C serial dependency.
        v8f a0 = cinit;
        v8f a1 = {};
#pragma unroll
        for (int j = 0; j < 16; j += 2) {
            const float* bp0 = xt + (size_t)(4 * j) * HW;
            const float* bp1 = xt + (size_t)(4 * j + 4) * HW;
            v2f b0, b1;
            b0.x = bp0[xoff];           // B rows K=4j+dl / +1, col N=pl
            b0.y = (bp0 + HW)[xoff];    // (shared voffset VGPR)
            b1.x = bp1[xoff];
            b1.y = (bp1 + HW)[xoff];
            a0 = __builtin_amdgcn_wmma_f32_16x16x4_f32(
                false, wa[j],     false, b0, (short)0, a0, false, false);
            a1 = __builtin_amdgcn_wmma_f32_16x16x4_f32(
                false, wa[j + 1], false, b1, (short)0, a1, false, false);
        }
        v8f acc;
#pragma unroll
        for (int r = 0; r < 8; ++r) acc[r] = a0[r] + a1[r];   // pk-addable

        // Gumbel-max over this lane's 8 k-rows. u+1e-20 can never be denormal,
        // so raw v_log_f32 (log2) is safe: ln(x) = ln2 * log2(x).
        const float* ut = urow + (size_t)t * 16 * KDIM;       // scalar advance
        v4f u0 = *(const v4f*)(ut + uoff);
        v4f u1 = *(const v4f*)(ut + uoff + 4u);
        float us[8] = {u0.x, u0.y, u0.z, u0.w, u1.x, u1.y, u1.z, u1.w};

        float best  = -3.4e38f;
        int   bestk = 0;
#pragma unroll
        for (int r = 0; r < 8; ++r) {
            float l1    = __builtin_amdgcn_logf(us[r] + EPSF);  // log2(u+eps)
            float inner = fmaf(-LN2F, l1, EPSF);                // -ln(u+eps)+eps > 0
            float g     = -LN2F * __builtin_amdgcn_logf(inner); // -ln(inner)
            float v     = acc[r] + g;
            int   k     = (int)ko + r;
            if (v > best) { best = v; bestk = k; }  // ascending k => first-max wins
        }
        // Combine half-wave k-groups (lane L <-> L^16); lowest k on ties.
        float ob  = __shfl_xor(best, 16, 32);
        int   obk = __shfl_xor(bestk, 16, 32);
        if (ob > best || (ob == best && obk < bestk)) { best = ob; bestk = obk; }

        // Emit palette color: lower half writes ch0/ch1, upper half writes ch2.
        float* ot = orow + t * 16;                 // scalar advance
        if (hi == 0u) {
            ot[pl]           = palette[bestk * 3 + 0];
            ot[pl + HW]      = palette[bestk * 3 + 1];
        } else {
            ot[pl + 2u * HW] = palette[bestk * 3 + 2];
        }
    }
}

extern "C" void kernel_launch(void* const* d_in, const int* in_sizes, int n_in,
                              void* d_out, int out_size, void* d_ws, size_t ws_size,
                              hipStream_t stream) {
    const float* x       = (const float*)d_in[0];
    const float* conv_w  = (const float*)d_in[1];
    const float* conv_b  = (const float*)d_in[2];
    const float* palette = (const float*)d_in[3];
    const float* unif    = (const float*)d_in[4];
    float* out           = (float*)d_out;

    // pixels = B*H*W = unif elements / K; 128 px per wave, 8 waves per block.
    const int pixels = in_sizes[4] / KDIM;          // 2,097,152
    const int blocks = pixels / (16 * TPW * 8);     // 2048
    quantizer_wmma_kernel<<<blocks, 256, 0, stream>>>(x, conv_w, conv_b,
                                                      palette, unif, out);
}